// GCN_71614284693688
// MI455X (gfx1250) — compile-verified
//
#include <hip/hip_runtime.h>
#include <hip/hip_bf16.h>

typedef __attribute__((ext_vector_type(2))) float v2f;
typedef __attribute__((ext_vector_type(8))) float v8f;

__device__ __forceinline__ void atomicAddF(float* p, float v) {
  // unsafeAtomicAdd forces the native global_atomic_add_f32 path (no CAS loop),
  // which is required to stay bandwidth-bound on the 205M-atomic scatter.
  unsafeAtomicAdd(p, v);
}

// ---------------------------------------------------------------------------
// deg[dst] += 1 per edge (float counts; self-loop folded into dinv kernel)
// ---------------------------------------------------------------------------
__global__ __launch_bounds__(256) void gcn_count_deg(const int* __restrict__ dst,
                                                     float* __restrict__ deg, int E) {
  int t = blockIdx.x * 256 + threadIdx.x;
  if (t < E) atomicAddF(&deg[dst[t]], 1.0f);
}

// dinv[n] = (deg[n] + 1)^-0.5   (in place; +1 = self loop, always > 0)
__global__ __launch_bounds__(256) void gcn_make_dinv(float* __restrict__ deg, int N) {
  int t = blockIdx.x * 256 + threadIdx.x;
  if (t < N) deg[t] = rsqrtf(deg[t] + 1.0f);
}

// ---------------------------------------------------------------------------
// C[M x 128] = A[M x 128] @ B[128 x 128], exact fp32 via V_WMMA_F32_16X16X4_F32
// One block (8 waves) per 16-row tile; wave w owns columns [16w, 16w+16).
// B (64KB) + A tile (8KB) staged in LDS.
// ---------------------------------------------------------------------------
__global__ __launch_bounds__(256) void gcn_gemm_f32(const float* __restrict__ A,
                                                    const float* __restrict__ B,
                                                    float* __restrict__ C, int M) {
  __shared__ float sB[128 * 128];
  __shared__ float sA[16 * 128];
  const int tid  = threadIdx.x;
  const int wave = tid >> 5;       // 0..7 -> column tile
  const int lane = tid & 31;
  const int m0   = blockIdx.x * 16;

  // stage weights: 16384 floats / 256 threads = 16 float4 each, coalesced
  {
    const float4* Bv = (const float4*)B;
    float4* sBv = (float4*)sB;
#pragma unroll
    for (int i = 0; i < 16; ++i) sBv[i * 256 + tid] = Bv[i * 256 + tid];
  }
  // stage A tile (16 x 128) with row clamp (stores are guarded later)
  {
    float4* sAv = (float4*)sA;
    const int rt = tid >> 5;       // 0..7
    const int ct = tid & 31;       // float4 column 0..31
    int r1 = m0 + rt;      if (r1 >= M) r1 = M - 1;
    int r2 = m0 + rt + 8;  if (r2 >= M) r2 = M - 1;
    sAv[rt * 32 + ct]       = ((const float4*)(A + (size_t)r1 * 128))[ct];
    sAv[(rt + 8) * 32 + ct] = ((const float4*)(A + (size_t)r2 * 128))[ct];
  }
  __syncthreads();

  v8f acc = {};
  const int mrow = lane & 15;            // A row / C col within tile
  const int kh   = (lane >> 4) << 1;     // 0 or 2 (K sub-offset per half-wave)
  const int col  = wave * 16 + mrow;     // B/C column
#pragma unroll
  for (int k0 = 0; k0 < 128; k0 += 4) {
    const int kb = k0 + kh;
    v2f a = *(const v2f*)&sA[mrow * 128 + kb];     // A[m][kb], A[m][kb+1]
    v2f b;
    b.x = sB[kb * 128 + col];                      // B[kb][col]
    b.y = sB[(kb + 1) * 128 + col];                // B[kb+1][col]
    acc = __builtin_amdgcn_wmma_f32_16x16x4_f32(false, a, false, b,
                                                (short)0, acc, false, false);
  }

  const int r0 = (lane >> 4) * 8;        // C: VGPR i -> row i + 8*(lane/16)
#pragma unroll
  for (int i = 0; i < 8; ++i) {
    int r = m0 + r0 + i;
    if (r < M) C[(size_t)r * 128 + col] = acc[i];
  }
}

// ---------------------------------------------------------------------------
// One wave per edge: agg[dst] += h[src] * dinv[src]*dinv[dst]  (128 feats,
// float4 per lane, native fp32 atomics)
// ---------------------------------------------------------------------------
__global__ __launch_bounds__(256) void gcn_scatter_edges(const float* __restrict__ h,
                                                         const int* __restrict__ srcI,
                                                         const int* __restrict__ dstI,
                                                         const float* __restrict__ dinv,
                                                         float* __restrict__ agg, int E) {
  int t = blockIdx.x * 256 + threadIdx.x;
  int e = t >> 5;
  if (e >= E) return;
  int lane = t & 31;
  int s = srcI[e];
  int d = dstI[e];
  float norm = dinv[s] * dinv[d];
  float4 hv = ((const float4*)(h + (size_t)s * 128))[lane];
  float* o = agg + (size_t)d * 128 + lane * 4;
  atomicAddF(o + 0, hv.x * norm);
  atomicAddF(o + 1, hv.y * norm);
  atomicAddF(o + 2, hv.z * norm);
  atomicAddF(o + 3, hv.w * norm);
}

// agg = relu(agg + h * dinv^2 + bias)   (self-loop term + bias + activation)
__global__ __launch_bounds__(256) void gcn_post(const float* __restrict__ h,
                                                const float* __restrict__ dinv,
                                                const float* __restrict__ bias,
                                                float* __restrict__ agg, int N) {
  int t = blockIdx.x * 256 + threadIdx.x;
  int n = t >> 5;
  if (n >= N) return;
  int lane = t & 31;
  float di = dinv[n];
  float sl = di * di;
  float4 hv = ((const float4*)(h + (size_t)n * 128))[lane];
  float4 av = ((const float4*)(agg + (size_t)n * 128))[lane];
  float4 bv = ((const float4*)bias)[lane];
  float4 r;
  r.x = fmaxf(fmaf(hv.x, sl, av.x) + bv.x, 0.0f);
  r.y = fmaxf(fmaf(hv.y, sl, av.y) + bv.y, 0.0f);
  r.z = fmaxf(fmaf(hv.z, sl, av.z) + bv.z, 0.0f);
  r.w = fmaxf(fmaf(hv.w, sl, av.w) + bv.w, 0.0f);
  ((float4*)(agg + (size_t)n * 128))[lane] = r;
}

// pooled[batch[n]] += h[n];  cnt[batch[n]] += 1  (one wave per node)
__global__ __launch_bounds__(256) void gcn_pool(const float* __restrict__ h,
                                                const int* __restrict__ batch,
                                                float* __restrict__ pooled,
                                                float* __restrict__ cnt, int N) {
  int t = blockIdx.x * 256 + threadIdx.x;
  int n = t >> 5;
  if (n >= N) return;
  int lane = t & 31;
  int g = batch[n];
  float4 hv = ((const float4*)(h + (size_t)n * 128))[lane];
  float* p = pooled + (size_t)g * 128 + lane * 4;
  atomicAddF(p + 0, hv.x);
  atomicAddF(p + 1, hv.y);
  atomicAddF(p + 2, hv.z);
  atomicAddF(p + 3, hv.w);
  if (lane == 0) atomicAddF(&cnt[g], 1.0f);
}

// out[g] = (pooled[g] . Wlin) / max(cnt[g],1) + blin   (wave per graph,
// wave32 shuffle reduction)
__global__ __launch_bounds__(256) void gcn_head(const float* __restrict__ pooled,
                                                const float* __restrict__ cnt,
                                                const float* __restrict__ Wlin,
                                                const float* __restrict__ blin,
                                                float* __restrict__ out, int G) {
  int t = blockIdx.x * 256 + threadIdx.x;
  int g = t >> 5;
  if (g >= G) return;
  int lane = t & 31;
  float4 pv = ((const float4*)(pooled + (size_t)g * 128))[lane];
  float4 wv = ((const float4*)Wlin)[lane];
  float s = pv.x * wv.x + pv.y * wv.y + pv.z * wv.z + pv.w * wv.w;
#pragma unroll
  for (int off = 16; off > 0; off >>= 1) s += __shfl_xor(s, off, 32);
  if (lane == 0) out[g] = s / fmaxf(cnt[g], 1.0f) + blin[0];
}

// ---------------------------------------------------------------------------
extern "C" void kernel_launch(void* const* d_in, const int* in_sizes, int n_in,
                              void* d_out, int out_size, void* d_ws, size_t ws_size,
                              hipStream_t stream) {
  const float* x     = (const float*)d_in[0];
  const int*   src   = (const int*)d_in[1];
  const int*   dst   = (const int*)d_in[2];
  const int*   batch = (const int*)d_in[3];
  // d_in[4] = n_graphs scalar; out_size == n_graphs
  const float* W1    = (const float*)d_in[5];
  const float* b1    = (const float*)d_in[6];
  const float* W2    = (const float*)d_in[7];
  const float* b2    = (const float*)d_in[8];
  const float* Wlin  = (const float*)d_in[9];
  const float* blin  = (const float*)d_in[10];
  float*       out   = (float*)d_out;

  const int N = in_sizes[0] / 128;   // nodes (IN_DIM == HID == 128)
  const int E = in_sizes[1];         // edges
  const int G = out_size;            // graphs

  // workspace layout (256B aligned chunks)
  size_t Nal = ((size_t)N + 63) & ~(size_t)63;
  float* dinv   = (float*)d_ws;                 // N (deg, then dinv in place)
  float* bufA   = dinv + Nal;                   // N*128 (GEMM output h)
  float* bufB   = bufA + (size_t)N * 128;       // N*128 (agg / layer output)
  float* pooled = bufB + (size_t)N * 128;       // G*128
  float* cnt    = pooled + (size_t)G * 128;     // G
  (void)ws_size; (void)n_in;

  const int nodeWaveBlocks = (int)(((size_t)N * 32 + 255) / 256);
  const int edgeWaveBlocks = (int)(((size_t)E * 32 + 255) / 256);
  const int gemmBlocks     = (N + 15) / 16;

  // degrees -> dinv (shared by both layers)
  hipMemsetAsync(dinv, 0, Nal * sizeof(float), stream);
  gcn_count_deg<<<(E + 255) / 256, 256, 0, stream>>>(dst, dinv, E);
  gcn_make_dinv<<<(N + 255) / 256, 256, 0, stream>>>(dinv, N);

  // ---- layer 1 ----
  gcn_gemm_f32<<<gemmBlocks, 256, 0, stream>>>(x, W1, bufA, N);
  hipMemsetAsync(bufB, 0, (size_t)N * 128 * sizeof(float), stream);
  gcn_scatter_edges<<<edgeWaveBlocks, 256, 0, stream>>>(bufA, src, dst, dinv, bufB, E);
  gcn_post<<<nodeWaveBlocks, 256, 0, stream>>>(bufA, dinv, b1, bufB, N);  // bufB = h1

  // ---- layer 2 ----
  gcn_gemm_f32<<<gemmBlocks, 256, 0, stream>>>(bufB, W2, bufA, N);
  hipMemsetAsync(bufB, 0, (size_t)N * 128 * sizeof(float), stream);
  gcn_scatter_edges<<<edgeWaveBlocks, 256, 0, stream>>>(bufA, src, dst, dinv, bufB, E);
  gcn_post<<<nodeWaveBlocks, 256, 0, stream>>>(bufA, dinv, b2, bufB, N);  // bufB = h2

  // ---- mean pool + linear head ----
  hipMemsetAsync(pooled, 0, ((size_t)G * 128 + G) * sizeof(float), stream);
  gcn_pool<<<nodeWaveBlocks, 256, 0, stream>>>(bufB, batch, pooled, cnt, N);
  gcn_head<<<(int)(((size_t)G * 32 + 255) / 256), 256, 0, stream>>>(pooled, cnt,
                                                                    Wlin, blin, out, G);
}